// DynamicSSM_LowRank_34033320853879
// MI455X (gfx1250) — compile-verified
//
#include <hip/hip_runtime.h>

// MI455X (gfx1250) wave32 WMMA implementation.
// Pipeline:
//   prep_T   : S = VC^H * UB (4x4 complex), T = UC * S (256x4 complex)  -> ws
//   prep_AM  : A[i][j] = UA*VA^H (rank4), M[i][j] = T*VB^H (rank4),
//              stored interleaved float4 {Ar,Ai,Mr,Mi} indexed [k=i][n=j] -> ws
//   ssm_main : per-frequency fused H-build + complex matmul via
//              v_wmma_f32_16x16x4_f32 on a 16-row real stack [x_r ; x_i].

typedef float v2f __attribute__((ext_vector_type(2)));
typedef float v8f __attribute__((ext_vector_type(8)));

#define NFREQ 256
#define XROWS 16
#define XPITCH 260  // 256 + 4 pad: 4-bank shift per row -> conflict-free ds_load_b64

// ------------------------------------------------------------------
// P1: S = VC^H UB (4x4 complex); T = UC * S (256x4 complex)
// ------------------------------------------------------------------
__global__ __launch_bounds__(256) void prep_T(
    const float* __restrict__ UB_r, const float* __restrict__ UB_i,
    const float* __restrict__ UC_r, const float* __restrict__ UC_i,
    const float* __restrict__ VC_r, const float* __restrict__ VC_i,
    float* __restrict__ Tr, float* __restrict__ Ti) {
  __shared__ float Sr[16], Si[16];
  const int t = threadIdx.x;
  if (t < 16) {
    const int p = t >> 2, q = t & 3;
    float sr = 0.f, si = 0.f;
    for (int j = 0; j < NFREQ; ++j) {
      float vcr = VC_r[j * 4 + p], vci = VC_i[j * 4 + p];
      float ubr = UB_r[j * 4 + q], ubi = UB_i[j * 4 + q];
      sr += vcr * ubr + vci * ubi;  // conj(VC) * UB : real
      si += vcr * ubi - vci * ubr;  //                 imag
    }
    Sr[t] = sr;
    Si[t] = si;
  }
  __syncthreads();
  const int i = t;
  #pragma unroll
  for (int q = 0; q < 4; ++q) {
    float tr = 0.f, ti = 0.f;
    #pragma unroll
    for (int p = 0; p < 4; ++p) {
      float ucr = UC_r[i * 4 + p], uci = UC_i[i * 4 + p];
      float sr = Sr[p * 4 + q], si = Si[p * 4 + q];
      tr += ucr * sr - uci * si;
      ti += ucr * si + uci * sr;
    }
    Tr[i * 4 + q] = tr;
    Ti[i * 4 + q] = ti;
  }
}

// ------------------------------------------------------------------
// P2: per (i,j): A = UA VA^H (rank 4), M = T VB^H (rank 4)
//     AM[i*256 + j] = {Ar, Ai, Mr, Mi}
// ------------------------------------------------------------------
__global__ __launch_bounds__(256) void prep_AM(
    const float* __restrict__ UA_r, const float* __restrict__ UA_i,
    const float* __restrict__ VA_r, const float* __restrict__ VA_i,
    const float* __restrict__ VB_r, const float* __restrict__ VB_i,
    const float* __restrict__ Tr, const float* __restrict__ Ti,
    float4* __restrict__ AM) {
  const int i = blockIdx.x, j = threadIdx.x;
  float ar = 0.f, ai = 0.f, mr = 0.f, mi = 0.f;
  #pragma unroll
  for (int r = 0; r < 4; ++r) {
    float ur = UA_r[i * 4 + r], ui = UA_i[i * 4 + r];
    float vr = VA_r[j * 4 + r], vi = VA_i[j * 4 + r];
    ar += ur * vr + ui * vi;  // UA * conj(VA)
    ai += ui * vr - ur * vi;
    float tr = Tr[i * 4 + r], ti = Ti[i * 4 + r];
    float wr = VB_r[j * 4 + r], wi = VB_i[j * 4 + r];
    mr += tr * wr + ti * wi;  // T * conj(VB)
    mi += ti * wr - tr * wi;
  }
  AM[i * NFREQ + j] = make_float4(ar, ai, mr, mi);
}

// ------------------------------------------------------------------
// Main: one block per frequency f. X = [x_r(b=0..7); x_i(b=0..7)] (16x256)
// staged in LDS. 8 waves, each owns 2 column tiles of 16.
// Per K-step of 4: build 4x16 tiles of H_r/H_i on the fly from AM + e^{-i ang},
// then two f32 WMMAs:  P1 += X*H_r , P2 += X*H_i.
// out_r = top(P1) - bottom(P2); out_i = top(P2) + bottom(P1) (shfl_xor 16).
// ------------------------------------------------------------------
__global__ __launch_bounds__(256) void ssm_main(
    const float* __restrict__ x_real, const float* __restrict__ x_imag,
    const float4* __restrict__ AM, float* __restrict__ out) {
  __shared__ __align__(16) float X[XROWS * XPITCH];
  const int f = blockIdx.x;
  const int t = threadIdx.x;

  // Stage X tile (16 rows x 256 cols) into LDS, b128 stores.
  {
    const int row = t >> 4;           // 0..15
    const int kb = (t & 15) * 16;     // 16-float chunk
    const float* src = (row < 8)
        ? (x_real + (size_t)row * NFREQ * NFREQ + (size_t)f * NFREQ + kb)
        : (x_imag + (size_t)(row - 8) * NFREQ * NFREQ + (size_t)f * NFREQ + kb);
    float* dst = &X[row * XPITCH + kb];
    const float4* s4 = (const float4*)src;
    #pragma unroll
    for (int m = 0; m < 4; ++m) {
      float4 v = s4[m];
      *((float4*)(dst + 4 * m)) = v;
    }
  }
  __syncthreads();

  const int wave = t >> 5;
  const int lane = t & 31;
  const int mrow = lane & 15;          // A-operand row / tile-local column N
  const int ksel = (lane >> 4) * 2;    // 0 for lanes 0-15, 2 for lanes 16-31

  const float ang = 6.28318530718f * (float)f * (1.0f / 256.0f);
  const float c = __cosf(ang);  // e^{-i ang} = c - i s
  const float s = __sinf(ang);

  for (int jt = wave; jt < 16; jt += 8) {
    const int n = jt * 16 + mrow;      // global output column for this lane
    v8f accR = {0.f, 0.f, 0.f, 0.f, 0.f, 0.f, 0.f, 0.f};
    v8f accI = {0.f, 0.f, 0.f, 0.f, 0.f, 0.f, 0.f, 0.f};
    const float* xrow = &X[mrow * XPITCH + ksel];

    #pragma unroll 4
    for (int k0 = 0; k0 < NFREQ; k0 += 4) {
      // A operand: X[mrow, k0+ksel .. +1]  (VGPR0=K0/K2, VGPR1=K1/K3 layout)
      v2f a = *(const v2f*)(xrow + k0);
      const int ka = k0 + ksel;
      float4 e0 = AM[ka * NFREQ + n];        // (k = ka,   n)
      float4 e1 = AM[(ka + 1) * NFREQ + n];  // (k = ka+1, n)

      // H = M / (1 - A * e),  e = c - i*s  ->  d = 1 - (Ar c + Ai s) - i(Ai c - Ar s)
      float zr0 = e0.x * c + e0.y * s;
      float zi0 = e0.y * c - e0.x * s;
      float dr0 = 1.0f - zr0, di0 = -zi0;
      float inv0 = __builtin_amdgcn_rcpf(dr0 * dr0 + di0 * di0);
      float hr0 = (e0.z * dr0 + e0.w * di0) * inv0;
      float hi0 = (e0.w * dr0 - e0.z * di0) * inv0;

      float zr1 = e1.x * c + e1.y * s;
      float zi1 = e1.y * c - e1.x * s;
      float dr1 = 1.0f - zr1, di1 = -zi1;
      float inv1 = __builtin_amdgcn_rcpf(dr1 * dr1 + di1 * di1);
      float hr1 = (e1.z * dr1 + e1.w * di1) * inv1;
      float hi1 = (e1.w * dr1 - e1.z * di1) * inv1;

      v2f br; br.x = hr0; br.y = hr1;   // B operand rows K0/K2 (VGPR0), K1/K3 (VGPR1)
      v2f bi; bi.x = hi0; bi.y = hi1;

      accR = __builtin_amdgcn_wmma_f32_16x16x4_f32(
          false, a, false, br, (short)0, accR, false, false);
      accI = __builtin_amdgcn_wmma_f32_16x16x4_f32(
          false, a, false, bi, (short)0, accI, false, false);
    }

    // Combine halves. D layout: VGPR g holds rows g (lanes 0-15) and 8+g (16-31).
    #pragma unroll
    for (int g = 0; g < 8; ++g) {
      float p1 = accR[g];
      float p2 = accI[g];
      float s1 = __shfl_xor(p1, 16, 32);  // bottom half of accR -> lanes 0-15
      float s2 = __shfl_xor(p2, 16, 32);  // bottom half of accI -> lanes 0-15
      if (lane < 16) {
        float orr = p1 - s2;  // Xr*Hr - Xi*Hi
        float oii = p2 + s1;  // Xr*Hi + Xi*Hr
        size_t idx = ((size_t)g * NFREQ + f) * NFREQ + (size_t)n;
        ((float2*)out)[idx] = make_float2(orr, oii);  // complex64 interleaved
      }
    }
  }
}

// ------------------------------------------------------------------
extern "C" void kernel_launch(void* const* d_in, const int* in_sizes, int n_in,
                              void* d_out, int out_size, void* d_ws, size_t ws_size,
                              hipStream_t stream) {
  (void)in_sizes; (void)n_in; (void)out_size; (void)ws_size;
  const float* x_real = (const float*)d_in[0];
  const float* x_imag = (const float*)d_in[1];
  const float* UA_r = (const float*)d_in[2];
  const float* UA_i = (const float*)d_in[3];
  const float* VA_r = (const float*)d_in[4];
  const float* VA_i = (const float*)d_in[5];
  const float* UB_r = (const float*)d_in[6];
  const float* UB_i = (const float*)d_in[7];
  const float* VB_r = (const float*)d_in[8];
  const float* VB_i = (const float*)d_in[9];
  const float* UC_r = (const float*)d_in[10];
  const float* UC_i = (const float*)d_in[11];
  const float* VC_r = (const float*)d_in[12];
  const float* VC_i = (const float*)d_in[13];

  // workspace: AM table (1 MiB) then T (8 KiB)
  float4* AM = (float4*)d_ws;
  float* Tr = (float*)d_ws + (size_t)NFREQ * NFREQ * 4;
  float* Ti = Tr + NFREQ * 4;

  prep_T<<<1, 256, 0, stream>>>(UB_r, UB_i, UC_r, UC_i, VC_r, VC_i, Tr, Ti);
  prep_AM<<<NFREQ, 256, 0, stream>>>(UA_r, UA_i, VA_r, VA_i, VB_r, VB_i, Tr, Ti, AM);
  ssm_main<<<NFREQ, 256, 0, stream>>>(x_real, x_imag, AM, (float*)d_out);
}